// YoloLoss_53317724012960
// MI455X (gfx1250) — compile-verified
//
#include <hip/hip_runtime.h>
#include <math.h>

typedef __attribute__((ext_vector_type(2))) float v2f;
typedef __attribute__((ext_vector_type(8))) float v8f;

#define NBATCH 16
#define NCLS   80
#define CH     85
#define MAXB   64

__device__ __forceinline__ float bce_logits(float x, float t) {
    return fmaxf(x, 0.f) - x * t + log1pf(expf(-fabsf(x)));
}

// ---------------------------------------------------------------------------
// Phase A: per (level, n) compact the boxes where label_conf==1, in ascending
// cell-index order, capped at 64 (exactly mirrors stable top_k on binary conf).
// One wave per image.
// ---------------------------------------------------------------------------
__global__ __launch_bounds__(32)
void compact_boxes_kernel(const float* __restrict__ label, int HW3,
                          float4* __restrict__ boxesOut, int* __restrict__ countOut)
{
    const int n = blockIdx.x;
    const int lane = threadIdx.x;
    const float* lb = label + (size_t)n * HW3 * CH;
    float4* bo = boxesOut + n * MAXB;
    int base = 0;
    for (int start = 0; start < HW3 && base < MAXB; start += 32) {
        int cell = start + lane;
        float conf = 0.f;
        if (cell < HW3) conf = lb[(size_t)cell * CH + 4];
        unsigned mask = __builtin_amdgcn_ballot_w32(conf > 0.f);
        int rank = __popc(mask & ((1u << lane) - 1u));
        if (conf > 0.f) {
            int pos = base + rank;
            if (pos < MAXB) {
                const float* p = lb + (size_t)cell * CH;
                bo[pos] = make_float4(p[0], p[1], p[2], p[3]);
            }
        }
        base += __popc(mask);
    }
    if (lane == 0) countOut[n] = base < MAXB ? base : MAXB;
}

// ---------------------------------------------------------------------------
// Phase B: main loss pass. One thread per (anchor, cell). Thread index
// i = a*HW + hw so pred channel loads are coalesced across lanes.
// 80-class prob loss is wave-vote gated (labels are ~0.08% positive).
// Block reduction of (reg, conf, prob) via V_WMMA_F32_16X16X4_F32 with a
// column-indicator B matrix: category c accumulates into D[:, c].
// ---------------------------------------------------------------------------
__global__ __launch_bounds__(256)
void level_loss_kernel(const float* __restrict__ pred, const float* __restrict__ label,
                       int H, int W, float stride,
                       float aw0, float ah0, float aw1, float ah1, float aw2, float ah2,
                       const float4* __restrict__ boxes, const int* __restrict__ counts,
                       float* __restrict__ partials)
{
    const int HW = H * W;
    const int total = 3 * HW;
    const int n = blockIdx.y;
    const int tid = threadIdx.x;
    const int i = blockIdx.x * 256 + tid;

    __shared__ float4 sBoxes[MAXB];
    __shared__ float red[3 * 256];
    __shared__ float sred[32];

    const int count = counts[n];
    for (int j = tid; j < MAXB; j += 256) sBoxes[j] = boxes[n * MAXB + j];
    __syncthreads();

    float regA = 0.f, confA = 0.f, probA = 0.f;
    if (i < total) {
        const int a  = i / HW;
        const int hw = i - a * HW;
        const int w  = hw % W;
        const int h  = hw / W;
        const float aw = (a == 0) ? aw0 : (a == 1) ? aw1 : aw2;
        const float ah = (a == 0) ? ah0 : (a == 1) ? ah1 : ah2;

        // pred (N,255,H,W): channels strided by HW, lanes contiguous in hw.
        const float* pf = pred + (size_t)(n * 255 + a * CH) * HW + hw;
        // label (N,H,W,3,85): contiguous 85 floats per cell.
        const float* lb = label + (size_t)(n * HW + hw) * 255 + (size_t)a * CH;

        const float rx = pf[0];
        const float ry = pf[(size_t)1 * HW];
        const float rw = pf[(size_t)2 * HW];
        const float rh = pf[(size_t)3 * HW];
        const float rc = pf[(size_t)4 * HW];

        const float lx = lb[0], ly = lb[1], lw = lb[2], lh = lb[3], lc = lb[4];

        const float gx = (float)w, gy = (float)h;
        const float px = (1.f / (1.f + expf(-rx)) + gx) * stride;
        const float py = (1.f / (1.f + expf(-ry)) + gy) * stride;
        const float pw = expf(rw) * aw;
        const float ph = expf(rh) * ah;

        // max IoU against the valid true boxes (invalid entries contribute 0).
        const float pltx = px - pw * 0.5f, plty = py - ph * 0.5f;
        const float prbx = px + pw * 0.5f, prby = py + ph * 0.5f;
        const float pa = pw * ph;
        float maxiou = 0.f;
        for (int j = 0; j < count; ++j) {
            const float4 b = sBoxes[j];
            const float bhw = b.z * 0.5f, bhh = b.w * 0.5f;
            const float ltx = fmaxf(pltx, b.x - bhw);
            const float lty = fmaxf(plty, b.y - bhh);
            const float rbx = fminf(prbx, b.x + bhw);
            const float rby = fminf(prby, b.y + bhh);
            const float iw = fmaxf(rbx - ltx, 0.f);
            const float ih = fmaxf(rby - lty, 0.f);
            const float inter = iw * ih;
            const float uni = pa + b.z * b.w - inter;
            maxiou = fmaxf(maxiou, inter / fmaxf(uni, 1e-9f));
        }
        const float ignore = (maxiou < 0.5f) ? 1.f : 0.f;

        const float ox = lx / stride - gx;
        const float oy = ly / stride - gy;
        const float ow = lc * logf(lw / aw + 1e-7f);
        const float oh = lc * logf(lh / ah + 1e-7f);
        const float scale = 2.f - lw * lh * (1.f / (416.f * 416.f));

        regA = lc * scale * (bce_logits(rx, ox) + bce_logits(ry, oy))
             + lc * scale * 0.5f * ((rw - ow) * (rw - ow) + (rh - oh) * (rh - oh));

        const float bc = bce_logits(rc, lc);
        confA = lc * bc + (1.f - lc) * ignore * bc;

        // Wave-vote: skip 160 floats/cell of traffic when no lane has a positive.
        if (__builtin_amdgcn_ballot_w32(lc > 0.f) != 0u) {
            float ps = 0.f;
            for (int c = 0; c < NCLS; ++c) {
                const float rp = pf[(size_t)(5 + c) * HW];
                const float lp = lb[5 + c];
                ps += bce_logits(rp, lp);
            }
            probA = lc * ps;
        }
    }

    red[tid]       = regA;
    red[256 + tid] = confA;
    red[512 + tid] = probA;
    __syncthreads();

    // WMMA reduction: 12 chunks of 64 staged values; chunks 0-3 are reg,
    // 4-7 conf, 8-11 prob. B = indicator of column `cat`, so D[m][cat]
    // accumulates row sums of the right category.
    if (tid < 32) {
        v8f acc = {};
        const int m  = tid & 15;
        const bool hi = tid >= 16;
        #pragma unroll
        for (int chunk = 0; chunk < 12; ++chunk) {
            const int cat  = chunk >> 2;
            const int base = chunk * 64 + m * 4 + (hi ? 2 : 0);
            v2f a; a.x = red[base]; a.y = red[base + 1];
            const float bv = (m == cat) ? 1.f : 0.f;
            v2f b; b.x = bv; b.y = bv;
            acc = __builtin_amdgcn_wmma_f32_16x16x4_f32(false, a, false, b,
                                                        (short)0, acc, false, false);
        }
        float s = 0.f;
        #pragma unroll
        for (int r = 0; r < 8; ++r) s += acc[r];
        sred[tid] = s;   // lane L holds sum over its 8 D-rows of column L%16
    }
    __syncthreads();
    if (tid == 0) {
        float* P = partials + 3 * (blockIdx.y * gridDim.x + blockIdx.x);
        P[0] = sred[0] + sred[16];   // reg  = full column 0
        P[1] = sred[1] + sred[17];   // conf = full column 1
        P[2] = sred[2] + sred[18];   // prob = full column 2
    }
}

// ---------------------------------------------------------------------------
// Deterministic fixed-order final reduction (no float atomics).
// ---------------------------------------------------------------------------
__global__ __launch_bounds__(32)
void finalize_kernel(const float* __restrict__ partials, int nTriples, float invN,
                     float* __restrict__ out)
{
    const int lane = threadIdx.x;
    __shared__ float sr[32], sc[32], sp[32];
    float r = 0.f, c = 0.f, p = 0.f;
    for (int j = lane; j < nTriples; j += 32) {
        r += partials[3 * j + 0];
        c += partials[3 * j + 1];
        p += partials[3 * j + 2];
    }
    sr[lane] = r; sc[lane] = c; sp[lane] = p;
    __syncthreads();
    if (lane == 0) {
        float R = 0.f, C = 0.f, P = 0.f;
        for (int j = 0; j < 32; ++j) { R += sr[j]; C += sc[j]; P += sp[j]; }
        out[0] = R * invN;
        out[1] = C * invN;
        out[2] = P * invN;
    }
}

// ---------------------------------------------------------------------------
extern "C" void kernel_launch(void* const* d_in, const int* in_sizes, int n_in,
                              void* d_out, int out_size, void* d_ws, size_t ws_size,
                              hipStream_t stream)
{
    const float* preds[3]  = {(const float*)d_in[0], (const float*)d_in[1], (const float*)d_in[2]};
    const float* labels[3] = {(const float*)d_in[3], (const float*)d_in[4], (const float*)d_in[5]};
    float* out = (float*)d_out;

    static const int   HS[3]     = {52, 26, 13};
    static const float STR[3]    = {8.f, 16.f, 32.f};
    static const float ANC[3][6] = {
        {12.f, 16.f, 19.f, 36.f, 40.f, 28.f},
        {36.f, 75.f, 76.f, 55.f, 72.f, 146.f},
        {142.f, 110.f, 192.f, 243.f, 459.f, 401.f}};

    // Workspace layout: boxes [3][16][64] float4, counts [48] int, partials.
    char* ws = (char*)d_ws;
    float4* boxesBase   = (float4*)ws;
    int*    countsBase  = (int*)(ws + (size_t)3 * NBATCH * MAXB * sizeof(float4));
    float*  partialBase = (float*)(ws + (size_t)3 * NBATCH * MAXB * sizeof(float4) + 256);

    int bpn[3], pOff[3], off = 0;
    for (int l = 0; l < 3; ++l) {
        int HW3 = 3 * HS[l] * HS[l];
        bpn[l] = (HW3 + 255) / 256;
        pOff[l] = off;
        off += bpn[l] * NBATCH;
    }

    for (int l = 0; l < 3; ++l) {
        int HW3 = 3 * HS[l] * HS[l];
        compact_boxes_kernel<<<NBATCH, 32, 0, stream>>>(
            labels[l], HW3,
            boxesBase + (size_t)l * NBATCH * MAXB,
            countsBase + l * NBATCH);
    }
    for (int l = 0; l < 3; ++l) {
        dim3 grid(bpn[l], NBATCH);
        level_loss_kernel<<<grid, 256, 0, stream>>>(
            preds[l], labels[l], HS[l], HS[l], STR[l],
            ANC[l][0], ANC[l][1], ANC[l][2], ANC[l][3], ANC[l][4], ANC[l][5],
            boxesBase + (size_t)l * NBATCH * MAXB,
            countsBase + l * NBATCH,
            partialBase + 3 * pOff[l]);
    }
    finalize_kernel<<<1, 32, 0, stream>>>(partialBase, off, 1.f / (float)NBATCH, out);
}